// MultiHeadNodeAttention_67138928771101
// MI455X (gfx1250) — compile-verified
//
#include <hip/hip_runtime.h>
#include <hip/hip_bf16.h>
#include <math.h>

typedef __attribute__((ext_vector_type(2))) float v2f;
typedef __attribute__((ext_vector_type(8))) float v8f;

#define NN 20000
#define EE 320000
#define HH 4
#define N_IN 256
#define N_OUT 64
#define E_IN 64
#define E_OUT 64
#define ALPHA 0.2f

__device__ __forceinline__ float leaky_f(float x) { return x >= 0.f ? x : ALPHA * x; }

// float atomic max via signed/unsigned int ordering trick (valid with -inf init)
__device__ __forceinline__ void atomicMaxF(float* addr, float v) {
    if (v >= 0.f) atomicMax((int*)addr, __float_as_int(v));
    else          atomicMin((unsigned int*)addr, __float_as_uint(v));
}

// ---- wave32 WMMA: one wave computes a 16-row x 64-col strip, K-loop of 16x16x4.
// A: row-major MxK (lda), B: row-major KxN (ldb). Four 16x16 accumulators share
// each A fragment (register blocking: 4x less A traffic, 4 indep WMMA chains).
// A frag layout (ISA 7.12.2): lanes 0-15 row M=l, v[0]=K, v[1]=K+1; lanes 16-31 K+2,K+3.
// B frag: lanes 0-15 col N=l, rows K/K+1; lanes 16-31 rows K+2/K+3.
__device__ __forceinline__ void wmma_strip_f32(const float* __restrict__ A, int lda,
                                               const float* __restrict__ B, int ldb,
                                               int row0, int K, v8f acc[4]) {
    const int lane = threadIdx.x & 31;
    const int half = lane >> 4;
    const int l    = lane & 15;
    const float* arow  = A + (size_t)(row0 + l) * lda + 2 * half;
    const float* bbase = B + (size_t)(2 * half) * ldb + l;
    for (int k = 0; k < K; k += 4) {
        v2f a; a.x = arow[k]; a.y = arow[k + 1];
#pragma unroll
        for (int c = 0; c < 4; ++c) {
            v2f b;
            b.x = bbase[(size_t)k * ldb + 16 * c];
            b.y = bbase[(size_t)(k + 1) * ldb + 16 * c];
            acc[c] = __builtin_amdgcn_wmma_f32_16x16x4_f32(false, a, false, b,
                                                           (short)0, acc[c], false, false);
        }
    }
}

// C/D layout: VGPR r -> row = row0 + r + 8*half, col = col0 + l
__device__ __forceinline__ void wmma_store(v8f acc, float* __restrict__ C, int ldc,
                                           int row0, int col0, float scale) {
    const int lane = threadIdx.x & 31;
    const int half = lane >> 4;
    const int l    = lane & 15;
#pragma unroll
    for (int r = 0; r < 8; ++r)
        C[(size_t)(row0 + r + 8 * half) * ldc + col0 + l] = acc[r] * scale;
}

// ---------------- kernels ----------------

__global__ void init_neg_inf(float* __restrict__ p, int n) {
    int i = blockIdx.x * blockDim.x + threadIdx.x;
    if (i < n) p[i] = -INFINITY;
}

// v_e[h][f] = sum_o We[h][f][o] * ae_edge[h][o]   (folds g into the edge score)
__global__ void compute_ve(const float* __restrict__ We, const float* __restrict__ ae_edge,
                           float* __restrict__ v_e) {
    int t = threadIdx.x;              // 256 = H*64
    int hh = t >> 6, f = t & 63;
    float s = 0.f;
    for (int o = 0; o < 64; ++o)
        s += We[((size_t)hh * 64 + f) * 64 + o] * ae_edge[hh * 64 + o];
    v_e[t] = s;
}

// h[h][n][o] = node_fts @ Wn[h]   (WMMA, K = 256, 16x64 strip per wave)
__global__ void gemm_node_proj(const float* __restrict__ node_fts,
                               const float* __restrict__ Wn,
                               float* __restrict__ h_buf) {
    const int wave = threadIdx.x >> 5;
    const int row0 = (blockIdx.x * 8 + wave) * 16;
    if (row0 >= NN) return;                       // whole-wave guard: EXEC stays all-ones
    const int head = blockIdx.y;
    const float* B = Wn + (size_t)head * N_IN * N_OUT;
    v8f acc[4] = {{0.f}, {0.f}, {0.f}, {0.f}};
    wmma_strip_f32(node_fts, N_IN, B, N_OUT, row0, N_IN, acc);
    float* C = h_buf + (size_t)head * NN * N_OUT;
#pragma unroll
    for (int c = 0; c < 4; ++c)
        wmma_store(acc[c], C, N_OUT, row0, 16 * c, 1.0f);
}

// per-node score projections (4 dots of h row with attention vectors)
__global__ void compute_q(const float* __restrict__ h_buf,
                          const float* __restrict__ a_src, const float* __restrict__ a_dst,
                          const float* __restrict__ ae_src, const float* __restrict__ ae_dst,
                          float* __restrict__ q_src, float* __restrict__ q_dst,
                          float* __restrict__ qe_src, float* __restrict__ qe_dst) {
    int idx = blockIdx.x * blockDim.x + threadIdx.x;   // idx = h*N + n
    if (idx >= HH * NN) return;
    int hh = idx / NN;
    const float* hr = h_buf + (size_t)idx * 64;
    float s0 = 0.f, s1 = 0.f, s2 = 0.f, s3 = 0.f;
    for (int o = 0; o < 64; ++o) {
        float hv = hr[o];
        s0 += hv * a_src[hh * 64 + o];
        s1 += hv * a_dst[hh * 64 + o];
        s2 += hv * ae_src[hh * 64 + o];
        s3 += hv * ae_dst[hh * 64 + o];
    }
    q_src[idx] = s0; q_dst[idx] = s1; qe_src[idx] = s2; qe_dst[idx] = s3;
}

// edge scores + segment max (atomics)
__global__ void compute_scores(const int* __restrict__ edges, const float* __restrict__ edge_fts,
                               const float* __restrict__ v_e,
                               const float* __restrict__ q_src, const float* __restrict__ q_dst,
                               const float* __restrict__ qe_src, const float* __restrict__ qe_dst,
                               float* __restrict__ s_node, float* __restrict__ s_edge,
                               float* __restrict__ m_n, float* __restrict__ m_e) {
    int idx = blockIdx.x * blockDim.x + threadIdx.x;   // idx = h*E + e
    if (idx >= HH * EE) return;
    int hh = idx / EE, e = idx % EE;
    int src = edges[2 * e], dst = edges[2 * e + 1];
    const float* ef = edge_fts + (size_t)e * 64;
    const float* vv = v_e + hh * 64;
    float r = 0.f;
    for (int f = 0; f < 64; ++f) r += ef[f] * vv[f];
    float sn = leaky_f(q_src[hh * NN + src] + q_dst[hh * NN + dst]);
    float se = leaky_f(qe_src[hh * NN + src] + qe_dst[hh * NN + dst] + r);
    s_node[idx] = sn;
    s_edge[idx] = se;
    atomicMaxF(&m_n[hh * NN + dst], sn);
    atomicMaxF(&m_e[hh * NN + dst], se);
}

// exp(score - segmax) + segment denominators
__global__ void compute_exp(const int* __restrict__ edges,
                            const float* __restrict__ m_n, const float* __restrict__ m_e,
                            float* __restrict__ s_node, float* __restrict__ s_edge,
                            float* __restrict__ den_n, float* __restrict__ den_e) {
    int idx = blockIdx.x * blockDim.x + threadIdx.x;
    if (idx >= HH * EE) return;
    int hh = idx / EE, e = idx % EE;
    int dst = edges[2 * e + 1];
    float mn = m_n[hh * NN + dst]; if (mn == -INFINITY) mn = 0.f;
    float me = m_e[hh * NN + dst]; if (me == -INFINITY) me = 0.f;
    float exn = __expf(s_node[idx] - mn);
    float exe = __expf(s_edge[idx] - me);
    s_node[idx] = exn;
    s_edge[idx] = exe;
    atomicAdd(&den_n[hh * NN + dst], exn);
    atomicAdd(&den_e[hh * NN + dst], exe);
}

// normalize -> attn; accumulate per-head sum / sumsq for variance weights
__global__ void compute_attn(const int* __restrict__ edges,
                             const float* __restrict__ den_n, const float* __restrict__ den_e,
                             float* __restrict__ s_node, float* __restrict__ s_edge,
                             float* __restrict__ stats) {
    __shared__ float sh[4][256];
    int tid = threadIdx.x;
    int idx = blockIdx.x * 256 + tid;            // E % 256 == 0 -> block stays in one head
    int hh = idx / EE, e = idx % EE;
    int dst = edges[2 * e + 1];
    float an = s_node[idx] / (den_n[hh * NN + dst] + 1e-9f);
    float ae = s_edge[idx] / (den_e[hh * NN + dst] + 1e-9f);
    s_node[idx] = an;
    s_edge[idx] = ae;
    sh[0][tid] = an; sh[1][tid] = an * an; sh[2][tid] = ae; sh[3][tid] = ae * ae;
    __syncthreads();
    for (int s = 128; s > 0; s >>= 1) {
        if (tid < s)
            for (int q = 0; q < 4; ++q) sh[q][tid] += sh[q][tid + s];
        __syncthreads();
    }
    if (tid == 0)
        for (int q = 0; q < 4; ++q) atomicAdd(&stats[hh * 4 + q], sh[q][0]);
}

// weighted scatter: node_acc += attn_n * h[src]; edge_acc += attn_e * edge_fts[e]
__global__ void scatter_acc(const int* __restrict__ edges, const float* __restrict__ h_buf,
                            const float* __restrict__ edge_fts,
                            const float* __restrict__ s_node, const float* __restrict__ s_edge,
                            float* __restrict__ node_acc, float* __restrict__ edge_acc) {
    const int tid = threadIdx.x;
    const int hh = tid >> 6;       // 4 heads x 64 lanes
    const int o  = tid & 63;
    const int e0 = blockIdx.x * 32;
    for (int i = 0; i < 32; ++i) {
        int e = e0 + i;
        int src = edges[2 * e], dst = edges[2 * e + 1];
        float an = s_node[(size_t)hh * EE + e];
        float ae = s_edge[(size_t)hh * EE + e];
        atomicAdd(&node_acc[((size_t)hh * NN + dst) * 64 + o],
                  an * h_buf[((size_t)hh * NN + src) * 64 + o]);
        atomicAdd(&edge_acc[((size_t)hh * NN + dst) * 64 + o],
                  ae * edge_fts[(size_t)e * 64 + o]);
    }
}

// per-head variance -> softmax-over-heads weights
__global__ void compute_weights(const float* __restrict__ stats, float* __restrict__ w) {
    if (threadIdx.x != 0 || blockIdx.x != 0) return;
    const float Ef = (float)EE;
    float wn[4], we[4], sn = 0.f, se = 0.f;
    for (int hh = 0; hh < 4; ++hh) {
        float s = stats[hh * 4 + 0], sq = stats[hh * 4 + 1];
        float var = (sq - s * s / Ef) / (Ef - 1.f);
        var = fminf(fmaxf(var, -2.f), 2.f);
        wn[hh] = __expf(var); sn += wn[hh];
        s = stats[hh * 4 + 2]; sq = stats[hh * 4 + 3];
        var = (sq - s * s / Ef) / (Ef - 1.f);
        var = fminf(fmaxf(var, -2.f), 2.f);
        we[hh] = __expf(var); se += we[hh];
    }
    for (int hh = 0; hh < 4; ++hh) { w[hh] = wn[hh] / sn; w[4 + hh] = we[hh] / se; }
}

// out[:, 0:256] = node_acc scaled by w_n (transpose head-major -> interleaved concat)
__global__ void write_node_out(const float* __restrict__ node_acc,
                               const float* __restrict__ w, float* __restrict__ out) {
    int idx = blockIdx.x * blockDim.x + threadIdx.x;
    if (idx >= NN * 256) return;
    int n = idx >> 8, c = idx & 255;
    int hh = c >> 6, o = c & 63;
    out[(size_t)n * 512 + c] = node_acc[((size_t)hh * NN + n) * 64 + o] * w[hh];
}

// out[:, 256:512] = (edge_acc[h] @ We[h]) * w_e[h]   (WMMA, K = 64, fused scale+write)
__global__ void gemm_edge_out(const float* __restrict__ edge_acc,
                              const float* __restrict__ We,
                              const float* __restrict__ w, float* __restrict__ out) {
    const int wave = threadIdx.x >> 5;
    const int row0 = (blockIdx.x * 8 + wave) * 16;
    if (row0 >= NN) return;                       // whole-wave guard
    const int head = blockIdx.y;
    const float* A = edge_acc + (size_t)head * NN * E_IN;
    const float* B = We + (size_t)head * E_IN * E_OUT;
    float scale = w[4 + head];
    v8f acc[4] = {{0.f}, {0.f}, {0.f}, {0.f}};
    wmma_strip_f32(A, E_IN, B, E_OUT, row0, E_IN, acc);
    float* C = out + HH * N_OUT + head * E_OUT;   // column offset into concat
    const int ldc = HH * N_OUT + HH * E_OUT;      // 512
#pragma unroll
    for (int c = 0; c < 4; ++c)
        wmma_store(acc[c], C, ldc, row0, 16 * c, scale);
}

extern "C" void kernel_launch(void* const* d_in, const int* in_sizes, int n_in,
                              void* d_out, int out_size, void* d_ws, size_t ws_size,
                              hipStream_t stream) {
    (void)in_sizes; (void)n_in; (void)out_size; (void)ws_size;
    const float* node_fts = (const float*)d_in[0];
    const float* edge_fts = (const float*)d_in[1];
    const int*   edges    = (const int*)d_in[2];
    const float* Wn       = (const float*)d_in[3];
    const float* We       = (const float*)d_in[4];
    const float* a_src    = (const float*)d_in[5];
    const float* a_dst    = (const float*)d_in[6];
    const float* ae_src   = (const float*)d_in[7];
    const float* ae_dst   = (const float*)d_in[8];
    const float* ae_edge  = (const float*)d_in[9];
    float* out = (float*)d_out;
    float* ws  = (float*)d_ws;

    size_t off = 0;
    float* h_buf    = ws + off; off += (size_t)HH * NN * 64;
    float* node_acc = ws + off; off += (size_t)HH * NN * 64;
    float* edge_acc = ws + off; off += (size_t)HH * NN * 64;   // contiguous with node_acc
    float* s_node   = ws + off; off += (size_t)HH * EE;
    float* s_edge   = ws + off; off += (size_t)HH * EE;
    float* q_src    = ws + off; off += (size_t)HH * NN;
    float* q_dst    = ws + off; off += (size_t)HH * NN;
    float* qe_src   = ws + off; off += (size_t)HH * NN;
    float* qe_dst   = ws + off; off += (size_t)HH * NN;
    float* m_n      = ws + off; off += (size_t)HH * NN;
    float* m_e      = ws + off; off += (size_t)HH * NN;        // contiguous with m_n
    float* den_n    = ws + off; off += (size_t)HH * NN;
    float* den_e    = ws + off; off += (size_t)HH * NN;        // contiguous with den_n
    float* v_e      = ws + off; off += (size_t)HH * 64;
    float* stats    = ws + off; off += 16;
    float* w        = ws + off; off += 8;

    // per-call re-init (graph-capture safe)
    hipMemsetAsync(node_acc, 0, sizeof(float) * 2ull * HH * NN * 64, stream);
    hipMemsetAsync(den_n,    0, sizeof(float) * 2ull * HH * NN, stream);
    hipMemsetAsync(stats,    0, sizeof(float) * 16, stream);
    init_neg_inf<<<(2 * HH * NN + 255) / 256, 256, 0, stream>>>(m_n, 2 * HH * NN);

    compute_ve<<<1, 256, 0, stream>>>(We, ae_edge, v_e);

    dim3 gh((NN / 16 + 7) / 8, HH);               // 157 x 4, 8 waves/block, 16x64 strip/wave
    gemm_node_proj<<<gh, 256, 0, stream>>>(node_fts, Wn, h_buf);

    compute_q<<<(HH * NN + 255) / 256, 256, 0, stream>>>(
        h_buf, a_src, a_dst, ae_src, ae_dst, q_src, q_dst, qe_src, qe_dst);

    int tot = HH * EE;                            // 1,280,000 (divisible by 256)
    compute_scores<<<(tot + 255) / 256, 256, 0, stream>>>(
        edges, edge_fts, v_e, q_src, q_dst, qe_src, qe_dst, s_node, s_edge, m_n, m_e);
    compute_exp<<<(tot + 255) / 256, 256, 0, stream>>>(
        edges, m_n, m_e, s_node, s_edge, den_n, den_e);
    compute_attn<<<tot / 256, 256, 0, stream>>>(
        edges, den_n, den_e, s_node, s_edge, stats);

    scatter_acc<<<EE / 32, 256, 0, stream>>>(
        edges, h_buf, edge_fts, s_node, s_edge, node_acc, edge_acc);

    compute_weights<<<1, 1, 0, stream>>>(stats, w);
    write_node_out<<<(NN * 256 + 255) / 256, 256, 0, stream>>>(node_acc, w, out);
    gemm_edge_out<<<gh, 256, 0, stream>>>(edge_acc, We, w, out);
}